// SSIMLoss_23098334118003
// MI455X (gfx1250) — compile-verified
//
#include <hip/hip_runtime.h>

typedef float v2f __attribute__((ext_vector_type(2)));
typedef float v8f __attribute__((ext_vector_type(8)));

#define XS 29   // LDS stride for 32x28 input patches (+1 pad)
#define HS 33   // LDS stride for transposed H tiles   (+1 pad)

// 11-tap Gaussian, sigma=1.5, normalized (matches reference _gaussian_window)
__device__ __constant__ float c_G[11] = {
    0.00102838f, 0.00759875f, 0.03600077f, 0.10936070f, 0.21300560f,
    0.26601172f,
    0.21300560f, 0.10936070f, 0.03600077f, 0.00759875f, 0.00102838f
};

// One wave (32 threads) computes one 16x16 output tile of the SSIM map for one
// image, fully fused: stage padded input patch -> horizontal blur via
// WMMA f32 16x16x4 (input x Toeplitz weights) -> vertical blur via WMMA
// (Toeplitz weights x H) -> pointwise SSIM -> wave reduction -> 1 partial.
__global__ __launch_bounds__(32)
void ssim_tile_kernel(const float* __restrict__ pred,
                      const float* __restrict__ targ,
                      float* __restrict__ partial) {
    __shared__ float Xp[32 * XS];      // pred patch, rows rtop..rtop+31, cols cbase..cbase+27
    __shared__ float Xt[32 * XS];      // target patch
    __shared__ float HT[5][16 * HS];   // horizontal-blur result, transposed: [n][row]
    __shared__ float Gs[16];           // gaussian taps (11 valid, rest 0)

    const int lane = threadIdx.x;      // 0..31, wave32
    const int half = lane >> 4;        // 0 or 1
    const int lm   = lane & 15;

    const int c0    = blockIdx.x * 16;
    const int r0    = blockIdx.y * 16;
    const int cbase = c0 - 5;          // horizontal halo: cols cbase .. cbase+27
    const int rtop  = r0 - 8;          // vertical halo:   rows rtop  .. rtop+31
    const size_t ioff = (size_t)blockIdx.z * (512 * 512);

    if (lane < 16) Gs[lane] = (lane < 11) ? c_G[lane] : 0.0f;

    // ---- stage 32x28 zero-padded patches of pred/target into LDS ----
    for (int i = lane; i < 32 * 28; i += 32) {
        const int r  = i / 28;
        const int c  = i - r * 28;
        const int gr = rtop + r;
        const int gc = cbase + c;
        const bool ok = (gr >= 0) && (gr < 512) && (gc >= 0) && (gc < 512);
        const size_t g = ioff + (size_t)gr * 512 + (size_t)gc;
        Xp[r * XS + c] = ok ? pred[g] : 0.0f;
        Xt[r * XS + c] = ok ? targ[g] : 0.0f;
    }
    __syncthreads();

    // ---- per-lane Toeplitz weight operands, shared by both passes ----
    // wq[q][v] = w[(4q + 2*half + v) - lm], zero outside [0,10].
    // Serves as B (K x N, banded) in the horizontal pass and as A (M x K,
    // banded) in the vertical pass — identical per-lane register values.
    v2f wq[7];
#pragma unroll
    for (int q = 0; q < 7; ++q) {
#pragma unroll
        for (int v = 0; v < 2; ++v) {
            const int idx = 4 * q + 2 * half + v - lm;
            const int ci  = idx < 15 ? idx : 15;
            wq[q][v] = (idx >= 0) ? Gs[ci] : 0.0f;
        }
    }

    // ---- horizontal blur: two 16-row tiles -> 32 H rows, 5 fields ----
#pragma unroll
    for (int t = 0; t < 2; ++t) {
        v8f hacc[5];
#pragma unroll
        for (int f = 0; f < 5; ++f) hacc[f] = (v8f){};

        const int row = t * 16 + lm;   // A-matrix row for this lane
#pragma unroll
        for (int q = 0; q < 7; ++q) {
            const int col = 4 * q + 2 * half;           // K-chunk base + half offset
            const float p0 = Xp[row * XS + col];
            const float p1 = Xp[row * XS + col + 1];
            const float t0 = Xt[row * XS + col];
            const float t1 = Xt[row * XS + col + 1];
            v2f a;
            a[0] = p0;      a[1] = p1;
            hacc[0] = __builtin_amdgcn_wmma_f32_16x16x4_f32(false, a, false, wq[q], (short)0, hacc[0], false, false);
            a[0] = t0;      a[1] = t1;
            hacc[1] = __builtin_amdgcn_wmma_f32_16x16x4_f32(false, a, false, wq[q], (short)0, hacc[1], false, false);
            a[0] = p0 * p0; a[1] = p1 * p1;
            hacc[2] = __builtin_amdgcn_wmma_f32_16x16x4_f32(false, a, false, wq[q], (short)0, hacc[2], false, false);
            a[0] = t0 * t0; a[1] = t1 * t1;
            hacc[3] = __builtin_amdgcn_wmma_f32_16x16x4_f32(false, a, false, wq[q], (short)0, hacc[3], false, false);
            a[0] = p0 * t0; a[1] = p1 * t1;
            hacc[4] = __builtin_amdgcn_wmma_f32_16x16x4_f32(false, a, false, wq[q], (short)0, hacc[4], false, false);
        }
        // D layout: lane holds D[m = t*16 + v + 8*half][n = lm]; store transposed.
#pragma unroll
        for (int f = 0; f < 5; ++f)
#pragma unroll
            for (int v = 0; v < 8; ++v)
                HT[f][lm * HS + t * 16 + 8 * half + v] = hacc[f][v];
    }
    __syncthreads();

    // ---- vertical blur: D = Wv(16x28 Toeplitz) x H(28x16), rbase = rtop+3 ----
    v8f vacc[5];
#pragma unroll
    for (int f = 0; f < 5; ++f) vacc[f] = (v8f){};

#pragma unroll
    for (int q = 0; q < 7; ++q) {
        const int kr = 3 + 4 * q + 2 * half;   // H row for this lane's B pair
#pragma unroll
        for (int f = 0; f < 5; ++f) {
            v2f b;
            b[0] = HT[f][lm * HS + kr];
            b[1] = HT[f][lm * HS + kr + 1];
            vacc[f] = __builtin_amdgcn_wmma_f32_16x16x4_f32(false, wq[q], false, b, (short)0, vacc[f], false, false);
        }
    }

    // ---- pointwise SSIM over this lane's 8 output pixels ----
    const float C1 = 1e-4f, C2 = 9e-4f;
    float s = 0.0f;
#pragma unroll
    for (int v = 0; v < 8; ++v) {
        const float mu1 = vacc[0][v], mu2 = vacc[1][v];
        const float bp2 = vacc[2][v], bt2 = vacc[3][v], bpt = vacc[4][v];
        const float mu1s = mu1 * mu1, mu2s = mu2 * mu2, m12 = mu1 * mu2;
        const float s1 = bp2 - mu1s, s2 = bt2 - mu2s, s12 = bpt - m12;
        const float num = (2.0f * m12 + C1) * (2.0f * s12 + C2);
        const float den = (mu1s + mu2s + C1) * (s1 + s2 + C2);
        s += num / den;
    }

    // ---- wave32 reduction, one partial per tile ----
#pragma unroll
    for (int off = 16; off > 0; off >>= 1) s += __shfl_xor(s, off, 32);
    if (lane == 0) {
        const int bidx = (blockIdx.z * gridDim.y + blockIdx.y) * gridDim.x + blockIdx.x;
        partial[bidx] = s;
    }
}

// Deterministic final reduction: fixed-order strided sums + LDS tree.
__global__ __launch_bounds__(256)
void ssim_finalize(const float* __restrict__ partial, float* __restrict__ out,
                   int n, float inv_npix) {
    __shared__ float red[256];
    float s = 0.0f;
    for (int i = threadIdx.x; i < n; i += 256) s += partial[i];
    red[threadIdx.x] = s;
    __syncthreads();
    for (int w = 128; w > 0; w >>= 1) {
        if ((int)threadIdx.x < w) red[threadIdx.x] += red[threadIdx.x + w];
        __syncthreads();
    }
    if (threadIdx.x == 0) out[0] = 1.0f - red[0] * inv_npix;
}

extern "C" void kernel_launch(void* const* d_in, const int* in_sizes, int n_in,
                              void* d_out, int out_size, void* d_ws, size_t ws_size,
                              hipStream_t stream) {
    const float* pred = (const float*)d_in[0];
    const float* targ = (const float*)d_in[1];
    float* partial = (float*)d_ws;          // 32*32*32 floats = 128 KB scratch

    dim3 grid(32, 32, 32);                   // (tile_c, tile_r, image)
    ssim_tile_kernel<<<grid, 32, 0, stream>>>(pred, targ, partial);
    ssim_finalize<<<1, 256, 0, stream>>>(partial, (float*)d_out, 32 * 32 * 32,
                                         1.0f / (32.0f * 512.0f * 512.0f));
}